// SpanRepresentation_29437705847574
// MI455X (gfx1250) — compile-verified
//
#include <hip/hip_runtime.h>
#include <hip/hip_bf16.h>

// ---------------------------------------------------------------------------
// Problem constants (from the reference)
// ---------------------------------------------------------------------------
#define HIDDEN   768
#define EMBED    50
#define TARGET   256
#define MAXLEN   512
#define SEQLEN   512
#define BATCH    2
#define MROWS    (SEQLEN * BATCH)       // 1024 rows of X (m = i*2 + b)
#define XCOLS    (2 * TARGET)           // 512: [Xs | Xe]

typedef __attribute__((ext_vector_type(2))) float v2f;
typedef __attribute__((ext_vector_type(4))) float v4f;
typedef __attribute__((ext_vector_type(8))) float v8f;

// ---------------------------------------------------------------------------
// Phase 1a: X[m, c] GEMM via V_WMMA_F32_16X16X4_F32.
//   X[m, c] = sum_h FS[b, i, h] * W[h + (c<256?0:768), c&255]
//   with m = i*2 + b  (matches (L, B) transpose in the reference).
// One wave computes one 16x16 C tile; K loop in steps of 4.
// ---------------------------------------------------------------------------
__global__ void __launch_bounds__(128)
xse_wmma_kernel(const float* __restrict__ FS, const float* __restrict__ W,
                float* __restrict__ X)
{
    const int lane = threadIdx.x & 31;
    const int wave = blockIdx.x * (blockDim.x >> 5) + (threadIdx.x >> 5);

    const int mt = wave >> 5;           // wave / 32  (0..63)
    const int nt = wave & 31;           // wave % 32  (0..31)
    const int m0 = mt * 16;
    const int c0 = nt * 16;

    const int half = lane >> 4;         // 0: K={k,k+1}  1: K={k+2,k+3}
    const int l15  = lane & 15;

    // A row pointer for this lane (feats transposed: row m -> FS[b=m&1, i=m>>1, :])
    const int m = m0 + l15;
    const float* __restrict__ Arow =
        FS + ((size_t)(m & 1) * SEQLEN + (size_t)(m >> 1)) * HIDDEN;

    // B column pointer (W row-major, stride TARGET); Xe columns use W rows 768..1535
    const int col  = c0 + l15;
    const int kofW = (c0 < TARGET) ? 0 : HIDDEN;
    const float* __restrict__ Bcol = W + (size_t)kofW * TARGET + (col & (TARGET - 1));

    v8f acc = {};
    #pragma unroll 4
    for (int k = 0; k < HIDDEN; k += 4) {
        const int ka = k + half * 2;
        v2f a;                              // A: two consecutive K per lane -> b64 load
        a.x = Arow[ka];
        a.y = Arow[ka + 1];
        v2f bf;                             // B: stride-TARGET dword loads
        bf.x = Bcol[(size_t)ka * TARGET];
        bf.y = Bcol[(size_t)(ka + 1) * TARGET];
        acc = __builtin_amdgcn_wmma_f32_16x16x4_f32(
            /*neg_a=*/false, a, /*neg_b=*/false, bf,
            /*c_mod=*/(short)0, acc, /*reuse_a=*/false, /*reuse_b=*/false);
    }

    // C/D layout: VGPR r -> M=r (lanes 0..15) / M=r+8 (lanes 16..31); N = lane&15
    const int   n     = l15;
    const int   rbase = half * 8;
    float* __restrict__ Cout = X + (size_t)m0 * XCOLS + c0;
    #pragma unroll
    for (int r = 0; r < 8; ++r)
        Cout[(size_t)(rbase + r) * XCOLS + n] = acc[r];
}

// ---------------------------------------------------------------------------
// Phase 1b: E[d, t] = b[t] + sum_k emb[d, k] * W[1536 + k, t]   (K=50, tiny)
// One block (256 threads) per row d; bias folded in here.
// ---------------------------------------------------------------------------
__global__ void __launch_bounds__(256)
e_kernel(const float* __restrict__ emb, const float* __restrict__ W,
         const float* __restrict__ bias, float* __restrict__ E)
{
    const int idx = blockIdx.x * 256 + threadIdx.x;      // 1024*256 threads
    const int d = idx >> 8;
    const int t = idx & (TARGET - 1);
    const float* __restrict__ er = emb + (size_t)d * EMBED;          // uniform per block
    const float* __restrict__ wc = W + (size_t)(2 * HIDDEN) * TARGET + t;
    float s = bias[t];
    #pragma unroll
    for (int k = 0; k < EMBED; ++k)
        s = fmaf(er[k], wc[(size_t)k * TARGET], s);
    E[idx] = s;
}

// ---------------------------------------------------------------------------
// Phase 2 (the 512 MB store-bound part):
//   out[((i*L + j)*2 + b)*256 + t] = Xs[i*2+b, t] + Xe[j*2+b, t] + E[j-i+512, t]
// One thread per float4; non-temporal b128 stores (output streamed once).
// ---------------------------------------------------------------------------
__global__ void __launch_bounds__(256)
bcast_add_kernel(const float* __restrict__ X, const float* __restrict__ E,
                 float* __restrict__ out)
{
    const size_t gid = (size_t)blockIdx.x * 256 + threadIdx.x;
    const int    c   = (int)(gid & 127);     // float4 slot within 512-float (i,j) row
    const size_t ij  = gid >> 7;
    const int    j   = (int)(ij & (SEQLEN - 1));
    const int    i   = (int)(ij >> 9);
    const int    pos = c << 2;               // 0..508
    const int    bb  = pos >> 8;             // batch
    const int    t   = pos & (TARGET - 1);

    const v4f xs = *(const v4f*)(X + ((size_t)(i * 2 + bb) * XCOLS) + t);
    const v4f xe = *(const v4f*)(X + ((size_t)(j * 2 + bb) * XCOLS) + TARGET + t);
    const v4f e  = *(const v4f*)(E + ((size_t)(j - i + MAXLEN) * TARGET) + t);

    const v4f r = xs + xe + e;

    __builtin_nontemporal_store(r, (v4f*)(out + (gid << 2)));
}

// ---------------------------------------------------------------------------
// Launch
// ---------------------------------------------------------------------------
extern "C" void kernel_launch(void* const* d_in, const int* in_sizes, int n_in,
                              void* d_out, int out_size, void* d_ws, size_t ws_size,
                              hipStream_t stream)
{
    const float* FS   = (const float*)d_in[0];   // (2, 512, 768)
    const float* W    = (const float*)d_in[1];   // (1586, 256)
    const float* bias = (const float*)d_in[2];   // (256,)
    const float* emb  = (const float*)d_in[3];   // (1024, 50)
    float* out = (float*)d_out;                  // (512, 512, 2, 256)

    float* X = (float*)d_ws;                                 // 1024*512 f32 = 2 MB
    float* E = (float*)((char*)d_ws + (size_t)MROWS * XCOLS * sizeof(float)); // 1 MB

    // Phase 1a: 64 * 32 = 2048 tiles, 4 waves (128 thr) per block -> 512 blocks
    xse_wmma_kernel<<<512, 128, 0, stream>>>(FS, W, X);

    // Phase 1b: 1024 rows * 256 cols / 256 = 1024 blocks
    e_kernel<<<1024, 256, 0, stream>>>(emb, W, bias, E);

    // Phase 2: 512*512*128 float4 threads / 256 = 131072 blocks
    bcast_add_kernel<<<131072, 256, 0, stream>>>(X, E, out);
}